// ContrastiveLoss_85366769975457
// MI455X (gfx1250) — compile-verified
//
#include <hip/hip_runtime.h>
#include <hip/hip_bf16.h>

typedef __attribute__((ext_vector_type(2))) float v2f;
typedef __attribute__((ext_vector_type(8))) float v8f;

// -------------------------------------------------------------------------
// Kernel 1: per-row squared norms  sq[i] = sum_k emb[i,k]^2
// -------------------------------------------------------------------------
__global__ void cl_rownorm_kernel(const float* __restrict__ emb,
                                  float* __restrict__ sq, int D) {
    const int i = blockIdx.x * blockDim.x + threadIdx.x;
    const float4* p = reinterpret_cast<const float4*>(emb + (size_t)i * D);
    float s = 0.0f;
    const int n4 = D >> 2;
    for (int k = 0; k < n4; ++k) {
        float4 v = p[k];
        s += v.x * v.x + v.y * v.y + v.z * v.z + v.w * v.w;
    }
    sq[i] = s;
}

// -------------------------------------------------------------------------
// Kernel 2: fused Gram-tile (WMMA fp32 16x16x4) + contrastive loss + reduce.
// One wave per 16x16 tile of the Gram matrix; only tiles with jt >= it.
//
// A-matrix layout (32-bit, 16x4, ISA 7.12.2): lane L<16 holds A[M=L][K=0..1]
// in v0..v1; lane L+16 holds A[L][K=2..3]. B (4x16) mirrors this with N=L.
// C/D: VGPR r holds (M = r + 8*(lane>=16), N = lane&15).
// -------------------------------------------------------------------------
__global__ void cl_tile_loss_kernel(const float* __restrict__ emb,
                                    const int* __restrict__ labels,
                                    const float* __restrict__ sq,
                                    float* __restrict__ partials,
                                    int D) {
    __shared__ float wsum[8];
    const int lane = threadIdx.x & 31;
    const int wave = threadIdx.x >> 5;
    const int it = blockIdx.y;
    const int jt = blockIdx.x * 8 + wave;

    float mysum = 0.0f;

    if (jt >= it) {                       // wave-uniform: EXEC all-1s inside
        const int i_base = it * 16;
        const int j_base = jt * 16;
        const int half = lane >> 4;       // 0: K=0,1   1: K=2,3
        const int l15  = lane & 15;
        const float* pa = emb + (size_t)(i_base + l15) * D + 2 * half;
        const float* pb = emb + (size_t)(j_base + l15) * D + 2 * half;

        v8f c = {};
        #pragma unroll 4
        for (int k0 = 0; k0 < D; k0 += 4) {
            v2f a = *reinterpret_cast<const v2f*>(pa + k0);
            v2f b = *reinterpret_cast<const v2f*>(pb + k0);
            // D = A(16x4) * B(4x16) + C   -> v_wmma_f32_16x16x4_f32
            c = __builtin_amdgcn_wmma_f32_16x16x4_f32(
                    /*neg_a=*/false, a, /*neg_b=*/false, b,
                    /*c_mod=*/(short)0, c, /*reuse_a=*/false, /*reuse_b=*/false);
        }

        // Apply contrastive loss per element of the 16x16 tile.
        const int   j   = j_base + l15;   // column (N)
        const float sqj = sq[j];
        const int   lj  = labels[j];
        #pragma unroll
        for (int r = 0; r < 8; ++r) {
            const int i = i_base + r + 8 * half;   // row (M)
            if (j > i) {                            // strict upper triangle
                float d2 = sq[i] + sqj - 2.0f * c[r];
                d2 = fmaxf(d2, 0.0f);
                const float dist = sqrtf(d2 + 1e-8f);
                float l;
                if (labels[i] == lj) {
                    l = 0.5f * dist * dist;
                } else {
                    const float t = fmaxf(1.0f - dist, 0.0f);
                    l = 0.5f * t * t;
                }
                mysum += l;
            }
        }
    }

    // wave32 butterfly reduction
    #pragma unroll
    for (int off = 16; off > 0; off >>= 1)
        mysum += __shfl_xor(mysum, off, 32);

    if (lane == 0) wsum[wave] = mysum;
    __syncthreads();
    if (threadIdx.x == 0) {
        float s = 0.0f;
        #pragma unroll
        for (int w = 0; w < 8; ++w) s += wsum[w];
        partials[(size_t)blockIdx.y * gridDim.x + blockIdx.x] = s;
    }
}

// -------------------------------------------------------------------------
// Kernel 3: deterministic final reduction of block partials -> scalar mean.
// -------------------------------------------------------------------------
__global__ void cl_reduce_kernel(const float* __restrict__ partials,
                                 float* __restrict__ out,
                                 int n, float inv_count) {
    __shared__ float buf[256];
    float s = 0.0f;
    for (int idx = threadIdx.x; idx < n; idx += 256) s += partials[idx];
    buf[threadIdx.x] = s;
    __syncthreads();
    #pragma unroll
    for (int off = 128; off > 0; off >>= 1) {
        if ((int)threadIdx.x < off) buf[threadIdx.x] += buf[threadIdx.x + off];
        __syncthreads();
    }
    if (threadIdx.x == 0) out[0] = buf[0] * inv_count;
}

// -------------------------------------------------------------------------
extern "C" void kernel_launch(void* const* d_in, const int* in_sizes, int n_in,
                              void* d_out, int out_size, void* d_ws, size_t ws_size,
                              hipStream_t stream) {
    const float* emb    = (const float*)d_in[0];   // (B, D) fp32
    const int*   labels = (const int*)d_in[1];     // (B,)

    const int B = in_sizes[1];                     // 4096
    const int D = in_sizes[0] / B;                 // 256

    float* sq       = (float*)d_ws;                // B floats
    float* partials = sq + B;                      // (nt * nt/8) floats

    const int nt = B / 16;                         // 16x16 tiles per dim (256)
    const int gx = nt / 8;                         // 8 tiles (waves) per block

    // 1) row norms
    cl_rownorm_kernel<<<B / 256, 256, 0, stream>>>(emb, sq, D);

    // 2) fused WMMA Gram tiles + loss
    dim3 grid(gx, nt, 1);
    cl_tile_loss_kernel<<<grid, 256, 0, stream>>>(emb, labels, sq, partials, D);

    // 3) final reduction: divide by C(B,2) + 1e-8 (matches reference mask sum)
    const double cnt = 0.5 * (double)B * (double)(B - 1);
    const float inv_count = (float)(1.0 / (cnt + 1e-8));
    cl_reduce_kernel<<<1, 256, 0, stream>>>(partials, (float*)d_out,
                                            gx * nt, inv_count);
}